// Net_23536420782419
// MI455X (gfx1250) — compile-verified
//
#include <hip/hip_runtime.h>
#include <hip/hip_bf16.h>
#include <math.h>

// GCN forward: x[100000,20] fp32, edge_index[2,6.4M], W1[20,16], b1[16], W2[16,2], b2[2]
// out[100000,2] = log_softmax over 2 classes.

#define NN 100000
#define NE 6400000
#define FIN 20
#define FH 16

typedef __attribute__((ext_vector_type(2))) float v2f;
typedef __attribute__((ext_vector_type(8))) float v8f;

// ---------------- degree / dinv ----------------

__global__ __launch_bounds__(256) void k_init_deg(float* __restrict__ deg) {
    int i = blockIdx.x * blockDim.x + threadIdx.x;
    if (i < NN) deg[i] = 1.0f;  // self-loop contributes 1 to every node
}

__global__ __launch_bounds__(256) void k_deg(const int* __restrict__ dst,
                                             float* __restrict__ deg) {
    int e = blockIdx.x * blockDim.x + threadIdx.x;
    if (e < NE) atomicAdd(&deg[dst[e]], 1.0f);
}

__global__ __launch_bounds__(256) void k_dinv(float* __restrict__ dinv) {
    int i = blockIdx.x * blockDim.x + threadIdx.x;
    if (i < NN) dinv[i] = 1.0f / sqrtf(dinv[i]);   // deg >= 1 always
}

// ---------------- layer-1 GEMM via WMMA f32 16x16x4 ----------------
// One wave per 16-row tile of x (100000 = 6250 * 16, exact).
// A (16x4 f32): lane<16 -> {K=k0,k0+1}, lane>=16 -> {K=k0+2,k0+3}, M = lane%16
// B (4x16 f32): same K split across halves, N = lane%16
// D (16x16 f32): VGPR r holds M = r + 8*(lane/16), N = lane%16
__global__ __launch_bounds__(256) void k_gemm1_wmma(const float* __restrict__ x,
                                                    const float* __restrict__ W1,
                                                    const float* __restrict__ dinv,
                                                    float* __restrict__ xw1,
                                                    float* __restrict__ agg1) {
    const int lane  = threadIdx.x & 31;
    const int wave  = (blockIdx.x * blockDim.x + threadIdx.x) >> 5;
    if (wave >= NN / 16) return;               // wave-uniform: EXEC stays all-ones
    const int half  = lane >> 4;               // 0 | 1
    const int idx16 = lane & 15;               // A row / B col / D col
    const int row0  = wave * 16;

    v8f acc = {};
#pragma unroll
    for (int k0 = 0; k0 < FIN; k0 += 4) {
        const int ka = k0 + 2 * half;
        v2f a, b;
        a.x = x[(row0 + idx16) * FIN + ka];
        a.y = x[(row0 + idx16) * FIN + ka + 1];
        b.x = W1[ka * FH + idx16];
        b.y = W1[(ka + 1) * FH + idx16];
        acc = __builtin_amdgcn_wmma_f32_16x16x4_f32(
            /*neg_a=*/false, a, /*neg_b=*/false, b,
            /*c_mod=*/(short)0, acc, /*reuse_a=*/false, /*reuse_b=*/false);
    }

#pragma unroll
    for (int r = 0; r < 8; ++r) {
        const int m = row0 + r + 8 * half;
        const float v = acc[r];
        const float d = dinv[m];
        xw1[m * FH + idx16]  = v;          // kept for edge gather
        agg1[m * FH + idx16] = d * d * v;  // self-loop term initializes the sum
    }
}

// ---------------- layer-1 edge scatter ----------------

__global__ __launch_bounds__(256) void k_scatter1(const int* __restrict__ src,
                                                  const int* __restrict__ dst,
                                                  const float* __restrict__ dinv,
                                                  const float* __restrict__ xw1,
                                                  float* __restrict__ agg1) {
    int e = blockIdx.x * blockDim.x + threadIdx.x;
    if (e >= NE) return;
    const int s = src[e];
    const int d = dst[e];
    const float w = dinv[s] * dinv[d];
    const float4* hs = reinterpret_cast<const float4*>(xw1 + s * FH); // 16B-aligned
    float* ad = agg1 + d * FH;
#pragma unroll
    for (int q = 0; q < 4; ++q) {
        float4 v = hs[q];
        atomicAdd(ad + 4 * q + 0, w * v.x);
        atomicAdd(ad + 4 * q + 1, w * v.y);
        atomicAdd(ad + 4 * q + 2, w * v.z);
        atomicAdd(ad + 4 * q + 3, w * v.w);
    }
}

// ---------------- bias + ReLU + layer-2 GEMM (16->2) + self-loop init ----------------

__global__ __launch_bounds__(256) void k_layer2(const float* __restrict__ agg1,
                                                const float* __restrict__ b1,
                                                const float* __restrict__ W2,
                                                const float* __restrict__ dinv,
                                                float* __restrict__ hw2,
                                                float* __restrict__ agg2) {
    int i = blockIdx.x * blockDim.x + threadIdx.x;
    if (i >= NN) return;
    float o0 = 0.0f, o1 = 0.0f;
#pragma unroll
    for (int f = 0; f < FH; ++f) {
        float h = fmaxf(agg1[i * FH + f] + b1[f], 0.0f);
        o0 = fmaf(h, W2[f * 2 + 0], o0);
        o1 = fmaf(h, W2[f * 2 + 1], o1);
    }
    const float d = dinv[i];
    const float d2 = d * d;
    hw2[i * 2 + 0] = o0;
    hw2[i * 2 + 1] = o1;
    agg2[i * 2 + 0] = d2 * o0;   // self-loop term
    agg2[i * 2 + 1] = d2 * o1;
}

// ---------------- layer-2 edge scatter ----------------

__global__ __launch_bounds__(256) void k_scatter2(const int* __restrict__ src,
                                                  const int* __restrict__ dst,
                                                  const float* __restrict__ dinv,
                                                  const float* __restrict__ hw2,
                                                  float* __restrict__ agg2) {
    int e = blockIdx.x * blockDim.x + threadIdx.x;
    if (e >= NE) return;
    const int s = src[e];
    const int d = dst[e];
    const float w = dinv[s] * dinv[d];
    const float2 v = *reinterpret_cast<const float2*>(hw2 + s * 2);
    atomicAdd(agg2 + d * 2 + 0, w * v.x);
    atomicAdd(agg2 + d * 2 + 1, w * v.y);
}

// ---------------- bias + log_softmax (2 classes) ----------------

__global__ __launch_bounds__(256) void k_logsoftmax(const float* __restrict__ agg2,
                                                    const float* __restrict__ b2,
                                                    float* __restrict__ out) {
    int i = blockIdx.x * blockDim.x + threadIdx.x;
    if (i >= NN) return;
    const float a = agg2[i * 2 + 0] + b2[0];
    const float b = agg2[i * 2 + 1] + b2[1];
    const float m = fmaxf(a, b);
    const float lse = m + logf(expf(a - m) + expf(b - m));
    out[i * 2 + 0] = a - lse;
    out[i * 2 + 1] = b - lse;
}

extern "C" void kernel_launch(void* const* d_in, const int* in_sizes, int n_in,
                              void* d_out, int out_size, void* d_ws, size_t ws_size,
                              hipStream_t stream) {
    const float* x  = (const float*)d_in[0];
    const int*   ei = (const int*)d_in[1];    // [2, NE]: row0 = src, row1 = dst
    const float* W1 = (const float*)d_in[2];
    const float* b1 = (const float*)d_in[3];
    const float* W2 = (const float*)d_in[4];
    const float* b2 = (const float*)d_in[5];
    const int* src = ei;
    const int* dst = ei + NE;

    // workspace layout (floats): dinv[NN] | xw1[16*NN] | agg1[16*NN] | hw2[2*NN] | agg2[2*NN]
    float* ws   = (float*)d_ws;
    float* dinv = ws;                    // doubles as deg accumulator first
    float* xw1  = dinv + NN;
    float* agg1 = xw1 + 16 * NN;
    float* hw2  = agg1 + 16 * NN;
    float* agg2 = hw2 + 2 * NN;

    const int node_blocks = (NN + 255) / 256;
    const int edge_blocks = (NE + 255) / 256;
    const int wmma_waves  = NN / 16;                      // 6250, exact
    const int wmma_blocks = (wmma_waves + 7) / 8;         // 8 waves per 256-thr block

    k_init_deg<<<node_blocks, 256, 0, stream>>>(dinv);
    k_deg<<<edge_blocks, 256, 0, stream>>>(dst, dinv);
    k_dinv<<<node_blocks, 256, 0, stream>>>(dinv);
    k_gemm1_wmma<<<wmma_blocks, 256, 0, stream>>>(x, W1, dinv, xw1, agg1);
    k_scatter1<<<edge_blocks, 256, 0, stream>>>(src, dst, dinv, xw1, agg1);
    k_layer2<<<node_blocks, 256, 0, stream>>>(agg1, b1, W2, dinv, hw2, agg2);
    k_scatter2<<<edge_blocks, 256, 0, stream>>>(src, dst, dinv, hw2, agg2);
    k_logsoftmax<<<node_blocks, 256, 0, stream>>>(agg2, b2, (float*)d_out);
}